// RandHashProj_6691559047459
// MI455X (gfx1250) — compile-verified
//
#include <hip/hip_runtime.h>

// Feature-hashing signed scatter-add:
//   out[b, o] = sum_{i : sel[i]==o} sign[i] * x[b, i]
// B=32 rows, OUT_FEAT=1024 buckets, N columns (1M in the reference).
//
// MI455X strategy (memory-bound: ~136 MB @ 23.3 TB/s ~= 5.9 us floor):
//  - Full 32x1024 f32 accumulator (128 KB) lives in LDS per workgroup --
//    only possible with CDNA5's 320 KB/WGP LDS -> x/sel/sign each read once.
//  - LDS layout acc[b][o]: bank index = sel % 64 (random) -> conflict-free-ish.
//    (acc[o][b] would map each b-step to only 2 banks -> 16-way conflicts.)
//  - int4/float4 b128 coalesced loads, native ds_add_f32 LDS atomics,
//    non-returning global_atomic_add_f32 flush (unsafeAtomicAdd).
//  - All 32-bit indexing: every offset < 2^31 (x spans 128 MB), avoids
//    64-bit SALU carry chains in the hot loop.
//  - WMMA deliberately unused: densifying the one-hot projection is a
//    ~66 GFLOP GEMM (~26 us on the matrix pipes) vs the 5.9 us stream.

#define NBATCH    32
#define OUT_FEATS 1024
#define ACC_ELEMS (NBATCH * OUT_FEATS)   // 32768 floats = 128 KB LDS
#define THREADS   1024                   // 32 wave32 waves per workgroup

__global__ void rhp_zero_kernel(float* __restrict__ out, int n) {
    int i = blockIdx.x * blockDim.x + threadIdx.x;
    if (i < n) out[i] = 0.0f;
}

__global__ __launch_bounds__(THREADS)
void rhp_scatter_kernel(const float* __restrict__ x,
                        const int*   __restrict__ sel,
                        const float* __restrict__ sgn,
                        float*       __restrict__ out,
                        int n) {
    extern __shared__ float acc[];  // ACC_ELEMS floats, dynamic LDS (128 KB)

    // Zero the LDS accumulator.
    for (int e = threadIdx.x; e < ACC_ELEMS; e += THREADS) acc[e] = 0.0f;
    __syncthreads();

    const int stride = (int)gridDim.x * THREADS * 4;   // <= 8.4M, fits int
    for (int i0 = ((int)blockIdx.x * THREADS + threadIdx.x) * 4;
         i0 < n; i0 += stride) {
        if (i0 + 4 <= n) {
            // Vector path: 4 consecutive columns per thread (16B-aligned,
            // fully coalesced across the wave32).
            const int4   s4 = *reinterpret_cast<const int4*>(sel + i0);
            const float4 g4 = *reinterpret_cast<const float4*>(sgn + i0);
            float* a0 = acc + s4.x;
            float* a1 = acc + s4.y;
            float* a2 = acc + s4.z;
            float* a3 = acc + s4.w;
            const float* xp = x + i0;
#pragma unroll 16
            for (int b = 0; b < NBATCH; ++b) {
                const float4 xv = *reinterpret_cast<const float4*>(xp);
                xp += n;
                const int row = b * OUT_FEATS;
                atomicAdd(a0 + row, g4.x * xv.x);   // -> ds_add_f32
                atomicAdd(a1 + row, g4.y * xv.y);
                atomicAdd(a2 + row, g4.z * xv.z);
                atomicAdd(a3 + row, g4.w * xv.w);
            }
        } else {
            // Scalar tail (only taken when n % 4 != 0, at most 3 columns).
            for (int i = i0; i < n; ++i) {
                const int   s = sel[i];
                const float g = sgn[i];
                const float* xp = x + i;
#pragma unroll 8
                for (int b = 0; b < NBATCH; ++b) {
                    atomicAdd(acc + b * OUT_FEATS + s, g * xp[0]);
                    xp += n;
                }
            }
        }
    }

    __syncthreads();

    // Flush block-local accumulator to global output. Non-returning native
    // f32 atomic adds; skip zeros (idle blocks / empty buckets).
    for (int e = threadIdx.x; e < ACC_ELEMS; e += THREADS) {
        const float v = acc[e];
        if (v != 0.0f) {
            unsafeAtomicAdd(out + e, v);            // -> global_atomic_add_f32
        }
    }
}

extern "C" void kernel_launch(void* const* d_in, const int* in_sizes, int n_in,
                              void* d_out, int out_size, void* d_ws, size_t ws_size,
                              hipStream_t stream) {
    (void)n_in; (void)d_ws; (void)ws_size;
    const float* x   = (const float*)d_in[0];   // (B, N) row-major f32
    const int*   sel = (const int*)  d_in[1];   // (N,) i32
    const float* sgn = (const float*)d_in[2];   // (N,) f32
    float*       out = (float*)d_out;           // (B, OUT_FEATS) f32

    const int n = in_sizes[1];                  // N (columns)

    // Zero output (harness poisons it; same stream -> ordered before scatter).
    rhp_zero_kernel<<<(out_size + 255) / 256, 256, 0, stream>>>(out, out_size);

    const int cols4 = (n + 3) / 4;
    int blocks = (cols4 + THREADS - 1) / THREADS;     // 245 for N = 1M
    if (blocks < 1) blocks = 1;
    if (blocks > 2048) blocks = 2048;                 // grid-stride covers rest

    const size_t ldsBytes = (size_t)ACC_ELEMS * sizeof(float);  // 128 KB
    rhp_scatter_kernel<<<blocks, THREADS, ldsBytes, stream>>>(x, sel, sgn, out, n);
}